// WarpDTI_24988119728886
// MI455X (gfx1250) — compile-verified
//
#include <hip/hip_runtime.h>
#include <math.h>

namespace {
constexpr int SDIM = 128;                 // H = W = D = 128
constexpr int HWD  = SDIM * SDIM * SDIM;  // 2^21 voxels per (batch, channel)
constexpr int NB   = 2;                   // batch

__device__ __forceinline__ int lin(int x, int y, int z) {
    return (x << 14) | (y << 7) | z;      // (x*128 + y)*128 + z, dims are 2^7
}
} // namespace

__global__ __launch_bounds__(256) void warp_dti_rot_kernel(
    const float* __restrict__ dti,   // (B, 6, H, W, D)
    const float* __restrict__ ddf,   // (B, 3, H, W, D)
    float* __restrict__ out)         // (B, 6, H, W, D)
{
    const int idx = blockIdx.x * blockDim.x + threadIdx.x;
    const int vox = idx & (HWD - 1);
    const int b   = idx >> 21;
    const int z   = vox & 127;
    const int y   = (vox >> 7) & 127;
    const int x   = (vox >> 14) & 127;

    const float* __restrict__ ddfb = ddf + (size_t)(b * 3) * HWD;
    const float* __restrict__ dtib = dti + (size_t)(b * 6) * HWD;

    // Warm caches for the DTI gather neighborhood: |displacement| <= ~2.1 voxels,
    // so the center cacheline covers most of the 8 trilinear taps.
    // gfx1250: lowers to global_prefetch_b8.
#pragma unroll
    for (int c = 0; c < 6; ++c)
        __builtin_prefetch((const void*)(dtib + c * HWD + vox), 0, 3);

    // ---- displacement at this voxel (coalesced, z fastest) ----
    const float u0c = ddfb[vox];
    const float u1c = ddfb[HWD + vox];
    const float u2c = ddfb[2 * HWD + vox];

    // ---- Jacobian of ddf + I  (jnp.gradient: central interior, one-sided at faces) ----
    // Branchless: clamp neighbor index, scale 1.0 at faces / 0.5 interior.
    const int xm = max(x - 1, 0), xp = min(x + 1, SDIM - 1);
    const int ym = max(y - 1, 0), yp = min(y + 1, SDIM - 1);
    const int zm = max(z - 1, 0), zp = min(z + 1, SDIM - 1);
    const float sx = (x == 0 || x == SDIM - 1) ? 1.0f : 0.5f;
    const float sy = (y == 0 || y == SDIM - 1) ? 1.0f : 0.5f;
    const float sz = (z == 0 || z == SDIM - 1) ? 1.0f : 0.5f;
    const int lxp = lin(xp, y, z), lxm = lin(xm, y, z);
    const int lyp = lin(x, yp, z), lym = lin(x, ym, z);
    const int lzp = lin(x, y, zp), lzm = lin(x, y, zm);

    float U[9];
#pragma unroll
    for (int i = 0; i < 3; ++i) {
        const float* __restrict__ f = ddfb + i * HWD;
        U[3 * i + 0] = sx * (f[lxp] - f[lxm]);
        U[3 * i + 1] = sy * (f[lyp] - f[lym]);
        U[3 * i + 2] = sz * (f[lzp] - f[lzm]);
    }
    U[0] += 1.0f; U[4] += 1.0f; U[8] += 1.0f;

    // ---- trilinear warp of the 6 DTI channels (clamped borders, unclamped frac) ----
    const float cx = (float)x + u0c;
    const float cy = (float)y + u1c;
    const float cz = (float)z + u2c;
    const float fx0 = floorf(cx), fy0 = floorf(cy), fz0 = floorf(cz);
    const float fx = cx - fx0, fy = cy - fy0, fz = cz - fz0;
    const int x0 = min(max((int)fx0, 0), SDIM - 1), x1 = min(max((int)fx0 + 1, 0), SDIM - 1);
    const int y0 = min(max((int)fy0, 0), SDIM - 1), y1 = min(max((int)fy0 + 1, 0), SDIM - 1);
    const int z0 = min(max((int)fz0, 0), SDIM - 1), z1 = min(max((int)fz0 + 1, 0), SDIM - 1);

    const float gx = 1.0f - fx, gy = 1.0f - fy, gz = 1.0f - fz;
    const float w000 = gx * gy * gz, w001 = gx * gy * fz;
    const float w010 = gx * fy * gz, w011 = gx * fy * fz;
    const float w100 = fx * gy * gz, w101 = fx * gy * fz;
    const float w110 = fx * fy * gz, w111 = fx * fy * fz;

    const int l000 = lin(x0, y0, z0), l001 = lin(x0, y0, z1);
    const int l010 = lin(x0, y1, z0), l011 = lin(x0, y1, z1);
    const int l100 = lin(x1, y0, z0), l101 = lin(x1, y0, z1);
    const int l110 = lin(x1, y1, z0), l111 = lin(x1, y1, z1);

    float d[6];
#pragma unroll
    for (int c = 0; c < 6; ++c) {
        const float* __restrict__ p = dtib + c * HWD;
        float acc =      w000 * p[l000];
        acc = fmaf(w001, p[l001], acc);
        acc = fmaf(w010, p[l010], acc);
        acc = fmaf(w011, p[l011], acc);
        acc = fmaf(w100, p[l100], acc);
        acc = fmaf(w101, p[l101], acc);
        acc = fmaf(w110, p[l110], acc);
        acc = fmaf(w111, p[l111], acc);
        d[c] = acc;
    }

    // ---- Newton polar iteration with determinant scaling: U -> R ----
    // U_{k+1} = 0.5*(z*U + inv((z*U)^T)),  z = |det U|^(-1/3).
    // Fused form: adj((zU)^T) = z^2 adj(U^T), det((zU)^T) = z^3 det U
    //   =>  inv((zU)^T) = adj(U^T) / (z * det U).
    // One cofactor set of U^T feeds both det (cofactor dot) and the inverse.
#pragma unroll
    for (int it = 0; it < 6; ++it) {
        // A = U^T (row-major a_ij = U[j][i])
        const float a00 = U[0], a01 = U[3], a02 = U[6];
        const float a10 = U[1], a11 = U[4], a12 = U[7];
        const float a20 = U[2], a21 = U[5], a22 = U[8];
        // adj(A) entries per the reference's inv3 cofactor formulas
        const float c00 = a11 * a22 - a12 * a21;
        const float c01 = a02 * a21 - a01 * a22;
        const float c02 = a01 * a12 - a02 * a11;
        const float c10 = a12 * a20 - a10 * a22;
        const float c11 = a00 * a22 - a02 * a20;
        const float c12 = a02 * a10 - a00 * a12;
        const float c20 = a10 * a21 - a11 * a20;
        const float c21 = a01 * a20 - a00 * a21;
        const float c22 = a00 * a11 - a01 * a10;
        const float detA = a00 * c00 + a01 * c10 + a02 * c20;   // = det(U)
        // zeta = |det|^(-1/3) via hw transcendentals: v_log_f32 + v_exp_f32.
        // (det ~ 1 for this input; Newton is self-correcting w.r.t. the scale.)
        const float zeta =
            __builtin_amdgcn_exp2f(-(1.0f / 3.0f) * __builtin_amdgcn_logf(fabsf(detA)));
        const float hz = 0.5f * zeta;                                // 0.5 * z
        const float hr = 0.5f * __builtin_amdgcn_rcpf(zeta * detA);  // 0.5 / (z*det)
        // U_new[i][j] = hz * U[i][j] + hr * adj(U^T)[i][j]
        U[0] = fmaf(hz, a00, hr * c00);
        U[1] = fmaf(hz, a10, hr * c01);
        U[2] = fmaf(hz, a20, hr * c02);
        U[3] = fmaf(hz, a01, hr * c10);
        U[4] = fmaf(hz, a11, hr * c11);
        U[5] = fmaf(hz, a21, hr * c12);
        U[6] = fmaf(hz, a02, hr * c20);
        U[7] = fmaf(hz, a12, hr * c21);
        U[8] = fmaf(hz, a22, hr * c22);
    }

    // ---- S = R^T D R  (D symmetric; R row-major in U) ----
    const float D00 = d[0], D01 = d[1], D11 = d[2], D02 = d[3], D12 = d[4], D22 = d[5];
    float t[9]; // t = D * R
#pragma unroll
    for (int j = 0; j < 3; ++j) {
        t[0 + j] = D00 * U[0 + j] + D01 * U[3 + j] + D02 * U[6 + j];
        t[3 + j] = D01 * U[0 + j] + D11 * U[3 + j] + D12 * U[6 + j];
        t[6 + j] = D02 * U[0 + j] + D12 * U[3 + j] + D22 * U[6 + j];
    }
    // S[i][l] = sum_j R[j][i] * t[j][l]; emit (0,0),(1,0),(1,1),(2,0),(2,1),(2,2)
    const float S00 = U[0] * t[0] + U[3] * t[3] + U[6] * t[6];
    const float S10 = U[1] * t[0] + U[4] * t[3] + U[7] * t[6];
    const float S11 = U[1] * t[1] + U[4] * t[4] + U[7] * t[7];
    const float S20 = U[2] * t[0] + U[5] * t[3] + U[8] * t[6];
    const float S21 = U[2] * t[1] + U[5] * t[4] + U[8] * t[7];
    const float S22 = U[2] * t[2] + U[5] * t[5] + U[8] * t[8];

    float* __restrict__ ob = out + (size_t)(b * 6) * HWD + vox;
    ob[0 * HWD] = S00;
    ob[1 * HWD] = S10;
    ob[2 * HWD] = S11;
    ob[3 * HWD] = S20;
    ob[4 * HWD] = S21;
    ob[5 * HWD] = S22;
}

extern "C" void kernel_launch(void* const* d_in, const int* in_sizes, int n_in,
                              void* d_out, int out_size, void* d_ws, size_t ws_size,
                              hipStream_t stream) {
    const float* dti = (const float*)d_in[0]; // (2, 6, 128, 128, 128) f32
    const float* ddf = (const float*)d_in[1]; // (2, 3, 128, 128, 128) f32
    float* out = (float*)d_out;               // (2, 6, 128, 128, 128) f32

    const int nvox = NB * HWD;                // 4,194,304 lanes, 1 voxel each
    dim3 block(256);                          // 8 wave32 waves per block
    dim3 grid(nvox / 256);
    warp_dti_rot_kernel<<<grid, block, 0, stream>>>(dti, ddf, out);
}